// OrderCourierGNN_22814866276963
// MI455X (gfx1250) — compile-verified
//
#include <hip/hip_runtime.h>
#include <hip/hip_bf16.h>
#include <math.h>

typedef __attribute__((ext_vector_type(2))) float v2f;
typedef __attribute__((ext_vector_type(8))) float v8f;

#define NEG_SLOPE 0.2f

// ---------------------------------------------------------------------------
// Float atomic max via the classic orderable-int trick.
// Positive values: signed int max; negative values: unsigned int min.
// Emits native global_atomic_max_i32 / global_atomic_min_u32 (no CAS loop).
// Init value must be -FLT_MAX.
// ---------------------------------------------------------------------------
__device__ inline void atomicMaxF(float* addr, float v) {
    if (v >= 0.0f) {
        atomicMax((int*)addr, __float_as_int(v));
    } else {
        atomicMin((unsigned int*)addr, (unsigned int)__float_as_int(v));
    }
}

// ---------------------------------------------------------------------------
// C[N x 64] = A[N x K] * W[K x 64], fp32, via V_WMMA_F32_16X16X4_F32.
// Block = 128 threads = 4 waves; wave w computes the 16x16 tile at
// (rowBase, 16*w). K must be a multiple of 4 (100 and 64 both are).
//
// ISA layouts (CDNA5 ISA §7.12.2), wave32:
//   A 16x4 :  lane<16: M=lane,  VGPR0=K0, VGPR1=K1 ; lane>=16: K2,K3
//   B 4x16 :  lane<16: N=lane,  VGPR0=K0, VGPR1=K1 ; lane>=16: K2,K3
//   C 16x16:  VGPR r : lanes 0-15 -> M=r, N=lane ; lanes 16-31 -> M=r+8
// ---------------------------------------------------------------------------
__global__ void gemm64_wmma(const float* __restrict__ A, const float* __restrict__ W,
                            float* __restrict__ C, int N, int K) {
    const int wave    = threadIdx.x >> 5;
    const int lane    = threadIdx.x & 31;
    const int l15     = lane & 15;
    const int kh      = lane >> 4;       // 0 or 1 -> K pair {0,1} or {2,3}
    const int colBase = wave * 16;
    const int rowBase = blockIdx.x * 16;

    int row = rowBase + l15;
    if (row >= N) row = N - 1;           // clamp: keeps EXEC fully on for WMMA

    v8f c = {};
    const int kIters = K >> 2;
    for (int it = 0; it < kIters; ++it) {
        const int k0 = (it << 2) + (kh << 1);     // even -> 8B aligned (K even)
        v2f a = *reinterpret_cast<const v2f*>(A + (size_t)row * K + k0);
        v2f b;
        b.x = W[(size_t)k0 * 64 + colBase + l15];
        b.y = W[(size_t)(k0 + 1) * 64 + colBase + l15];
        c = __builtin_amdgcn_wmma_f32_16x16x4_f32(false, a, false, b,
                                                  (short)0, c, false, false);
    }
#pragma unroll
    for (int r = 0; r < 8; ++r) {
        const int orow = rowBase + r + 8 * kh;
        if (orow < N) C[(size_t)orow * 64 + colBase + l15] = c[r];
    }
}

// ---------------------------------------------------------------------------
// Per-node attention logits: as[n] = h[n,:].a_src ; ad[n] = h[n,:].a_dst
// One wave32 per node; lane l covers features l and l+32.
// ---------------------------------------------------------------------------
__global__ void node_alpha_kernel(const float* __restrict__ h,
                                  const float* __restrict__ asrc,
                                  const float* __restrict__ adst,
                                  float* __restrict__ out_s,
                                  float* __restrict__ out_d, int N) {
    const int wave = threadIdx.x >> 5;
    const int lane = threadIdx.x & 31;
    const int n = blockIdx.x * 8 + wave;
    if (n >= N) return;
    const float h0 = h[(size_t)n * 64 + lane];
    const float h1 = h[(size_t)n * 64 + lane + 32];
    float s = h0 * asrc[lane] + h1 * asrc[lane + 32];
    float d = h0 * adst[lane] + h1 * adst[lane + 32];
#pragma unroll
    for (int o = 16; o > 0; o >>= 1) {
        s += __shfl_down(s, o, 32);
        d += __shfl_down(d, o, 32);
    }
    if (lane == 0) { out_s[n] = s; out_d[n] = d; }
}

// ---------------------------------------------------------------------------
// Init per-layer accumulators: m = -FLT_MAX, denom = 0, acc = 0.
// ---------------------------------------------------------------------------
__global__ void init_kernel(float* __restrict__ m, float* __restrict__ denom,
                            float* __restrict__ acc, int N) {
    const int i = blockIdx.x * blockDim.x + threadIdx.x;
    if (i < N) { m[i] = -3.402823466e38f; denom[i] = 0.0f; }
    if (i < N * 64) acc[i] = 0.0f;
}

// ---------------------------------------------------------------------------
// Pass A: segment max of leaky_relu(as[src] + ad[dst]) over dst.
// Edges e < E come from edge_index; e >= E are self loops (src=dst=e-E).
// ---------------------------------------------------------------------------
__global__ void edge_max_kernel(const long long* __restrict__ ei, int E, int N,
                                const float* __restrict__ as,
                                const float* __restrict__ ad,
                                float* __restrict__ m) {
    const int e = blockIdx.x * blockDim.x + threadIdx.x;
    if (e >= E + N) return;
    int s, d;
    if (e < E) { s = (int)ei[e]; d = (int)ei[E + e]; }
    else       { s = d = e - E; }
    float v = as[s] + ad[d];
    v = v > 0.0f ? v : NEG_SLOPE * v;
    atomicMaxF(m + d, v);
}

// ---------------------------------------------------------------------------
// Pass B: w = exp(e - m[dst]); denom[dst] += w; acc[dst,:] += w * h[src,:].
// One wave32 per edge: lane l owns features {2l, 2l+1} (coalesced 256B gather,
// contiguous 256B of f32 atomics per edge, resolved in L2).
// ---------------------------------------------------------------------------
__global__ void edge_accum_kernel(const long long* __restrict__ ei, int E, int N,
                                  const float* __restrict__ as,
                                  const float* __restrict__ ad,
                                  const float* __restrict__ m,
                                  const float* __restrict__ h,
                                  float* __restrict__ acc,
                                  float* __restrict__ denom) {
    const int wave = threadIdx.x >> 5;
    const int lane = threadIdx.x & 31;
    const int e = blockIdx.x * 8 + wave;
    if (e >= E + N) return;
    int s, d;
    if (e < E) { s = (int)ei[e]; d = (int)ei[E + e]; }
    else       { s = d = e - E; }
    float v = as[s] + ad[d];
    v = v > 0.0f ? v : NEG_SLOPE * v;
    const float w = expf(v - m[d]);
    if (lane == 0) atomicAdd(denom + d, w);
    v2f hv = *reinterpret_cast<const v2f*>(h + (size_t)s * 64 + 2 * lane);
    atomicAdd(acc + (size_t)d * 64 + 2 * lane,     w * hv.x);
    atomicAdd(acc + (size_t)d * 64 + 2 * lane + 1, w * hv.y);
}

// ---------------------------------------------------------------------------
// out[n,k] = acc[n,k] / (denom[n] + 1e-16) + bias[k]  (optional ReLU), in place
// ---------------------------------------------------------------------------
__global__ void normalize_kernel(float* __restrict__ acc,
                                 const float* __restrict__ denom,
                                 const float* __restrict__ bias,
                                 int N, int do_relu) {
    const int i = blockIdx.x * blockDim.x + threadIdx.x;
    if (i >= N * 64) return;
    const int n = i >> 6, k = i & 63;
    float v = acc[i] / (denom[n] + 1e-16f) + bias[k];
    if (do_relu) v = fmaxf(v, 0.0f);
    acc[i] = v;
}

// ---------------------------------------------------------------------------
// score[e] = sigmoid(dot(h2[src], h2[dst])) over ORIGINAL edges only.
// ---------------------------------------------------------------------------
__global__ void edge_score_kernel(const long long* __restrict__ ei, int E,
                                  const float* __restrict__ h,
                                  float* __restrict__ out) {
    const int wave = threadIdx.x >> 5;
    const int lane = threadIdx.x & 31;
    const int e = blockIdx.x * 8 + wave;
    if (e >= E) return;
    const int s = (int)ei[e];
    const int d = (int)ei[E + e];
    v2f a = *reinterpret_cast<const v2f*>(h + (size_t)s * 64 + 2 * lane);
    v2f b = *reinterpret_cast<const v2f*>(h + (size_t)d * 64 + 2 * lane);
    float p = a.x * b.x + a.y * b.y;
#pragma unroll
    for (int o = 16; o > 0; o >>= 1) p += __shfl_down(p, o, 32);
    if (lane == 0) out[e] = 1.0f / (1.0f + expf(-p));
}

// ---------------------------------------------------------------------------
// Orchestration. Inputs (setup_inputs order):
// 0:x 1:edge_index(int64) 2:edge_attr(unused) 3:W1 4:a1_src 5:a1_dst 6:b1
// 7:W2 8:a2_src 9:a2_dst 10:b2
// ---------------------------------------------------------------------------
extern "C" void kernel_launch(void* const* d_in, const int* in_sizes, int n_in,
                              void* d_out, int out_size, void* d_ws, size_t ws_size,
                              hipStream_t stream) {
    const float*     x   = (const float*)d_in[0];
    const long long* ei  = (const long long*)d_in[1];
    const float*     W1  = (const float*)d_in[3];
    const float*     a1s = (const float*)d_in[4];
    const float*     a1d = (const float*)d_in[5];
    const float*     b1  = (const float*)d_in[6];
    const float*     W2  = (const float*)d_in[7];
    const float*     a2s = (const float*)d_in[8];
    const float*     a2d = (const float*)d_in[9];
    const float*     b2  = (const float*)d_in[10];

    const int DH  = in_sizes[4];            // 64
    const int DIN = in_sizes[3] / DH;       // 100
    const int N   = in_sizes[0] / DIN;      // 50000
    const int E   = in_sizes[1] / 2;        // 800000
    const int Esl = E + N;

    float* bufA  = (float*)d_ws;                  // N*64  (GEMM output h)
    float* bufB  = bufA + (size_t)N * 64;         // N*64  (acc -> features)
    float* as    = bufB + (size_t)N * 64;         // N
    float* ad    = as + N;                        // N
    float* m     = ad + N;                        // N
    float* denom = m + N;                         // N

    const int rowTiles = (N + 15) / 16;
    const int gN64     = (N * 64 + 255) / 256;

    // ---- Layer 1: h1 = GAT(x; W1) , ReLU ----
    gemm64_wmma      <<<rowTiles,         128, 0, stream>>>(x, W1, bufA, N, DIN);
    node_alpha_kernel<<<(N + 7) / 8,      256, 0, stream>>>(bufA, a1s, a1d, as, ad, N);
    init_kernel      <<<gN64,             256, 0, stream>>>(m, denom, bufB, N);
    edge_max_kernel  <<<(Esl + 255)/256,  256, 0, stream>>>(ei, E, N, as, ad, m);
    edge_accum_kernel<<<(Esl + 7) / 8,    256, 0, stream>>>(ei, E, N, as, ad, m, bufA, bufB, denom);
    normalize_kernel <<<gN64,             256, 0, stream>>>(bufB, denom, b1, N, 1);

    // ---- Layer 2: h2 = GAT(h1; W2) ----
    gemm64_wmma      <<<rowTiles,         128, 0, stream>>>(bufB, W2, bufA, N, DH);
    node_alpha_kernel<<<(N + 7) / 8,      256, 0, stream>>>(bufA, a2s, a2d, as, ad, N);
    init_kernel      <<<gN64,             256, 0, stream>>>(m, denom, bufB, N);
    edge_max_kernel  <<<(Esl + 255)/256,  256, 0, stream>>>(ei, E, N, as, ad, m);
    edge_accum_kernel<<<(Esl + 7) / 8,    256, 0, stream>>>(ei, E, N, as, ad, m, bufA, bufB, denom);
    normalize_kernel <<<gN64,             256, 0, stream>>>(bufB, denom, b2, N, 0);

    // ---- Edge scoring over original edges ----
    edge_score_kernel<<<(E + 7) / 8,      256, 0, stream>>>(ei, E, bufB, (float*)d_out);
}